// GATLayerPrimitive_41807211659464
// MI455X (gfx1250) — compile-verified
//
#include <hip/hip_runtime.h>

typedef __attribute__((ext_vector_type(2))) float v2f;
typedef __attribute__((ext_vector_type(8))) float v8f;

#define N_NODES   100000
#define N_EDGES   1600000
#define IN_CH     128
#define OUT_CH    64
#define NEG_SLOPE 0.2f
#define LDS_STRIDE (IN_CH + 1)   // pad to avoid 16-way LDS bank conflicts

// ---------------------------------------------------------------------------
// float atomic max via int/uint ordering trick (init value must be -inf)
// ---------------------------------------------------------------------------
__device__ __forceinline__ void atomicMaxFloat(float* addr, float val) {
    if (val >= 0.0f) atomicMax((int*)addr, __float_as_int(val));
    else             atomicMin((unsigned int*)addr, __float_as_uint(val));
}

// ---------------------------------------------------------------------------
// 0) init: zero h (=d_out), m = -inf, denom = 0
// ---------------------------------------------------------------------------
__global__ void init_kernel(float* __restrict__ h, float* __restrict__ mx,
                            float* __restrict__ dn) {
    int tid = blockIdx.x * blockDim.x + threadIdx.x;
    if (tid < N_NODES * OUT_CH) h[tid] = 0.0f;
    if (tid < N_NODES) { mx[tid] = -INFINITY; dn[tid] = 0.0f; }
}

// ---------------------------------------------------------------------------
// 1) Wx = x @ W^T with V_WMMA_F32_16X16X4_F32.
//    Block = 256 threads = 8 waves; wave w computes rows [blk*128 + 16w, +16)
//    across all 64 output cols (4 accumulators of 16x16).
//    Out-of-range rows are CLAMPED (not zeroed) for the A loads: a garbage A
//    row only pollutes the same D row, and the store is guarded. This keeps
//    the inner loop free of EXEC-mask branches.
// ---------------------------------------------------------------------------
__global__ __launch_bounds__(256)
void gemm_wmma_f32(const float* __restrict__ x, const float* __restrict__ W,
                   float* __restrict__ Wx) {
    __shared__ float Wlds[OUT_CH * LDS_STRIDE];   // 64 x 129 floats ~ 33 KB

    // cooperative load of W[64][128] into padded LDS
    for (int i = threadIdx.x; i < OUT_CH * IN_CH; i += 256) {
        int r = i >> 7;          // /128
        int c = i & 127;
        Wlds[r * LDS_STRIDE + c] = W[i];
    }
    __syncthreads();

    const int wave = threadIdx.x >> 5;
    const int lane = threadIdx.x & 31;
    const int hh   = lane >> 4;          // 0: K pair {0,1}; 1: K pair {2,3}
    const int lid  = lane & 15;
    const int rowbase = blockIdx.x * 128 + wave * 16;

    // clamped row pointer for unconditional A loads
    int arow = rowbase + lid;
    if (arow > N_NODES - 1) arow = N_NODES - 1;
    const float* xrow = x + (size_t)arow * IN_CH + 2 * hh;

    v8f acc0 = {}, acc1 = {}, acc2 = {}, acc3 = {};

    for (int kb = 0; kb < IN_CH; kb += 4) {
        const int k = kb + 2 * hh;

        // A fragment (16x4 f32): lane(lid,hh) holds x[arow][k], x[arow][k+1]
        v2f a;
        a.x = xrow[kb];
        a.y = xrow[kb + 1];

        // B fragments (4x16 f32 = W^T tile): lane holds W[n][k], W[n][k+1]
        v2f b0, b1, b2, b3;
        {
            const float* w0 = &Wlds[(0  + lid) * LDS_STRIDE + k];
            const float* w1 = &Wlds[(16 + lid) * LDS_STRIDE + k];
            const float* w2 = &Wlds[(32 + lid) * LDS_STRIDE + k];
            const float* w3 = &Wlds[(48 + lid) * LDS_STRIDE + k];
            b0.x = w0[0]; b0.y = w0[1];
            b1.x = w1[0]; b1.y = w1[1];
            b2.x = w2[0]; b2.y = w2[1];
            b3.x = w3[0]; b3.y = w3[1];
        }

        acc0 = __builtin_amdgcn_wmma_f32_16x16x4_f32(false, a, false, b0, (short)0, acc0, false, false);
        acc1 = __builtin_amdgcn_wmma_f32_16x16x4_f32(false, a, false, b1, (short)0, acc1, false, false);
        acc2 = __builtin_amdgcn_wmma_f32_16x16x4_f32(false, a, false, b2, (short)0, acc2, false, false);
        acc3 = __builtin_amdgcn_wmma_f32_16x16x4_f32(false, a, false, b3, (short)0, acc3, false, false);
    }

    // D layout: VGPR j -> row = rowbase + j + 8*hh, col = tile*16 + lid
    #pragma unroll
    for (int j = 0; j < 8; ++j) {
        const int mrow = rowbase + j + 8 * hh;
        if (mrow < N_NODES) {
            float* out = Wx + (size_t)mrow * OUT_CH + lid;
            out[0]  = acc0[j];
            out[16] = acc1[j];
            out[32] = acc2[j];
            out[48] = acc3[j];
        }
    }
}

// ---------------------------------------------------------------------------
// 2) per-node scores: s_i = Wx . a[0:64], s_j = Wx . a[64:128]
//    one wave32 per node; lane covers channels (lane, lane+32)
// ---------------------------------------------------------------------------
__global__ __launch_bounds__(256)
void node_scores(const float* __restrict__ Wx, const float* __restrict__ avec,
                 float* __restrict__ sI, float* __restrict__ sJ) {
    const int node = (blockIdx.x * blockDim.x + threadIdx.x) >> 5;
    const int lane = threadIdx.x & 31;
    if (node >= N_NODES) return;

    const float* row = Wx + (size_t)node * OUT_CH;
    const float v0 = row[lane];
    const float v1 = row[lane + 32];

    float pi = v0 * avec[lane]      + v1 * avec[lane + 32];
    float pj = v0 * avec[64 + lane] + v1 * avec[96 + lane];

    #pragma unroll
    for (int off = 16; off > 0; off >>= 1) {
        pi += __shfl_xor(pi, off, 32);
        pj += __shfl_xor(pj, off, 32);
    }
    if (lane == 0) { sI[node] = pi; sJ[node] = pj; }
}

// ---------------------------------------------------------------------------
// 3) edge pass 1: e = leaky_relu(s_i[dst] + s_j[src]); segment max into m[dst]
// ---------------------------------------------------------------------------
__global__ void edge_max(const long long* __restrict__ ei,
                         const float* __restrict__ sI, const float* __restrict__ sJ,
                         float* __restrict__ eb, float* __restrict__ mx) {
    const int e = blockIdx.x * blockDim.x + threadIdx.x;
    if (e >= N_EDGES) return;
    const int src = (int)ei[e];
    const int dst = (int)ei[N_EDGES + e];
    float v = sI[dst] + sJ[src];
    v = (v > 0.0f) ? v : NEG_SLOPE * v;
    eb[e] = v;
    atomicMaxFloat(&mx[dst], v);
}

// ---------------------------------------------------------------------------
// 4) edge pass 2: exp(e - m[dst]); segment sum into denom[dst]
// ---------------------------------------------------------------------------
__global__ void edge_expsum(const long long* __restrict__ ei,
                            const float* __restrict__ mx,
                            float* __restrict__ eb, float* __restrict__ dn) {
    const int e = blockIdx.x * blockDim.x + threadIdx.x;
    if (e >= N_EDGES) return;
    const int dst = (int)ei[N_EDGES + e];
    const float ex = __expf(eb[e] - mx[dst]);
    eb[e] = ex;
    atomicAdd(&dn[dst], ex);
}

// ---------------------------------------------------------------------------
// 5) edge pass 3: h[dst] += (exp_e/denom[dst]) * Wx[src]
//    one wave32 per edge; lane handles channels (lane, lane+32)
// ---------------------------------------------------------------------------
__global__ __launch_bounds__(256)
void edge_scatter(const long long* __restrict__ ei, const float* __restrict__ eb,
                  const float* __restrict__ dn, const float* __restrict__ Wx,
                  float* __restrict__ h) {
    const long long tid = (long long)blockIdx.x * blockDim.x + threadIdx.x;
    const int e    = (int)(tid >> 5);
    const int lane = (int)(tid & 31);
    if (e >= N_EDGES) return;

    const int src = (int)ei[e];
    const int dst = (int)ei[N_EDGES + e];
    const float alpha = eb[e] / dn[dst];

    const float* wrow = Wx + (size_t)src * OUT_CH;
    float* hrow = h + (size_t)dst * OUT_CH;
    atomicAdd(&hrow[lane],      alpha * wrow[lane]);
    atomicAdd(&hrow[lane + 32], alpha * wrow[lane + 32]);
}

// ---------------------------------------------------------------------------
// 6) ELU in place on d_out
// ---------------------------------------------------------------------------
__global__ void elu_kernel(float* __restrict__ h) {
    const int tid = blockIdx.x * blockDim.x + threadIdx.x;
    if (tid >= N_NODES * OUT_CH) return;
    const float v = h[tid];
    h[tid] = (v > 0.0f) ? v : (__expf(v) - 1.0f);
}

// ---------------------------------------------------------------------------
extern "C" void kernel_launch(void* const* d_in, const int* in_sizes, int n_in,
                              void* d_out, int out_size, void* d_ws, size_t ws_size,
                              hipStream_t stream) {
    const float*     x   = (const float*)d_in[0];
    const long long* ei  = (const long long*)d_in[1];   // int64 [2, E]
    const float*     W   = (const float*)d_in[2];
    const float*     av  = (const float*)d_in[3];
    float*           out = (float*)d_out;

    // workspace layout (floats)
    float* ws = (float*)d_ws;
    float* Wx = ws;                              // N*64
    float* sI = Wx + (size_t)N_NODES * OUT_CH;   // N
    float* sJ = sI + N_NODES;                    // N
    float* mx = sJ + N_NODES;                    // N
    float* dn = mx + N_NODES;                    // N
    float* eb = dn + N_NODES;                    // E

    const int TB = 256;

    // 0) init h / m / denom
    init_kernel<<<(N_NODES * OUT_CH + TB - 1) / TB, TB, 0, stream>>>(out, mx, dn);

    // 1) WMMA GEMM: 128 rows per block
    gemm_wmma_f32<<<(N_NODES + 127) / 128, TB, 0, stream>>>(x, W, Wx);

    // 2) node scores: 1 wave per node, 8 nodes per block
    node_scores<<<(N_NODES + 7) / 8, TB, 0, stream>>>(Wx, av, sI, sJ);

    // 3-4) edge softmax passes
    edge_max<<<(N_EDGES + TB - 1) / TB, TB, 0, stream>>>(ei, sI, sJ, eb, mx);
    edge_expsum<<<(N_EDGES + TB - 1) / TB, TB, 0, stream>>>(ei, mx, eb, dn);

    // 5) scatter: 1 wave per edge, 8 edges per block
    edge_scatter<<<(N_EDGES + 7) / 8, TB, 0, stream>>>(ei, eb, dn, Wx, out);

    // 6) ELU
    elu_kernel<<<(N_NODES * OUT_CH + TB - 1) / TB, TB, 0, stream>>>(out);
}